// ChatGLM3Attention_11467562680663
// MI455X (gfx1250) — compile-verified
//
#include <hip/hip_runtime.h>
#include <hip/hip_bf16.h>
#include <math.h>

// ---- problem constants ----
#define NUM_HEADS   32
#define NUM_KV      2
#define HD          128
#define HIDDEN      4096
#define NQKV        4608            // (32 + 2*2) * 128
#define BB          2
#define SS          2048

typedef __attribute__((ext_vector_type(2))) float v2f;
typedef __attribute__((ext_vector_type(8))) float v8f;

// CDNA5 async global->LDS copy (ASYNCcnt-tracked, no VGPR staging).
// vdst = per-lane LDS byte offset, vaddr = per-lane 64-bit global address.
__device__ __forceinline__ void async_ld_b128(unsigned int lds_off,
                                              const float* gptr) {
  asm volatile("global_load_async_to_lds_b128 %0, %1, off"
               :: "v"(lds_off), "v"(gptr)
               : "memory");
}

// ---------------------------------------------------------------------------
// Kernel 1: QKV projection  C[M=4096, N=4608] = H[4096,4096] @ W[4096,4608]
// fp32 WMMA 16x16x4. Block = 128 threads (4 waves), tile = 64 rows x 64 cols.
// A chunks (64 x 32 K) double-buffered in LDS via async loads; pad-36 stride
// keeps the A-operand ds_load_b64 reads bank-conflict free.
// ---------------------------------------------------------------------------
#define LAS 36
__global__ __launch_bounds__(128) void qkv_gemm_kernel(
    const float* __restrict__ A, const float* __restrict__ W,
    float* __restrict__ C) {
  __shared__ float lds_a[2][64 * LAS];
  const int m_base = blockIdx.x * 64;
  const int n_base = blockIdx.y * 64;
  const int t    = threadIdx.x;
  const int wave = t >> 5;
  const int lane = t & 31;
  const int row_a = lane & 15;            // A-operand row / B-operand col
  const int koff  = (lane >> 4) << 1;     // 0 for lanes 0-15, 2 for 16-31
  const int n_w   = n_base + wave * 16 + row_a;

  const unsigned int lb0 = (unsigned int)(uintptr_t)&lds_a[0][0];
  const unsigned int lb1 = (unsigned int)(uintptr_t)&lds_a[1][0];

  // per-thread cooperative slice: 4 float4 (async b128) per chunk
  auto issue_chunk = [&](int k0, unsigned int lbase) {
#pragma unroll
    for (int i = 0; i < 4; ++i) {
      int f  = t + i * 128;
      int r  = f >> 3;          // 8 float4 per row
      int c4 = f & 7;
      async_ld_b128(lbase + (unsigned int)(r * LAS + c4 * 4) * 4u,
                    &A[(size_t)(m_base + r) * HIDDEN + k0 + c4 * 4]);
    }
  };

  v8f acc[4] = {};
  issue_chunk(0, lb0);

  for (int k0 = 0, it = 0; k0 < HIDDEN; k0 += 32, ++it) {
    const int cur = it & 1;
    const unsigned int lcur = cur ? lb1 : lb0;
    if (k0 + 32 < HIDDEN) {
      issue_chunk(k0 + 32, cur ? lb0 : lb1);
      asm volatile("s_wait_asynccnt 4" ::: "memory");  // current chunk done
    } else {
      asm volatile("s_wait_asynccnt 0" ::: "memory");
    }
    __syncthreads();

    const float* la = (const float*)(uintptr_t)0 + 0;  // (unused; keep idx math below)
#pragma unroll
    for (int kk = 0; kk < 8; ++kk) {
      const int kc = 4 * kk + koff;
      v2f b;
      b.x = W[(size_t)(k0 + kc    ) * NQKV + n_w];
      b.y = W[(size_t)(k0 + kc + 1) * NQKV + n_w];
#pragma unroll
      for (int mt = 0; mt < 4; ++mt) {
        v2f a = *reinterpret_cast<const v2f*>(
            &lds_a[cur][(mt * 16 + row_a) * LAS + kc]);
        acc[mt] = __builtin_amdgcn_wmma_f32_16x16x4_f32(
            false, a, false, b, (short)0, acc[mt], false, false);
      }
    }
    (void)la;
    __syncthreads();  // all waves done reading lds_a[cur] before it is refilled
  }

  const int row_off = (lane >> 4) * 8;   // C/D layout: lanes 16-31 hold M+8
#pragma unroll
  for (int mt = 0; mt < 4; ++mt)
#pragma unroll
    for (int j = 0; j < 8; ++j) {
      int r = m_base + mt * 16 + row_off + j;
      C[(size_t)r * NQKV + n_w] = acc[mt][j];
    }
}

// ---------------------------------------------------------------------------
// Kernel 2: RoPE in-place on Q (heads 0..31) and K (heads 32..33) of qkv ws.
// grid = (B*S, 34), block = 64 (one thread per rotation pair).
// ---------------------------------------------------------------------------
__global__ __launch_bounds__(64) void rope_kernel(
    const int* __restrict__ pos, float* __restrict__ qkv) {
  const int bs = blockIdx.x;            // 0 .. B*S-1
  const int h  = blockIdx.y;            // 0..31 => Q head, 32..33 => K head
  const int d  = threadIdx.x;           // 0..63
  const int s  = bs & (SS - 1);
  const float p = (float)pos[s];
  // inv_freq = 10000^(-d/64) = exp(-d/64 * ln 10000)
  const float inv_freq = __expf(-(float)d * (1.0f / 64.0f) * 9.210340371976184f);
  const float ang = p * inv_freq;
  const float c  = cosf(ang);
  const float sn = sinf(ang);
  const size_t base = (size_t)bs * NQKV +
                      (h < NUM_HEADS ? h * HD : NUM_HEADS * HD + (h - NUM_HEADS) * HD);
  const float x1 = qkv[base + d];
  const float x2 = qkv[base + d + 64];
  qkv[base + d]      = x1 * c - x2 * sn;
  qkv[base + d + 64] = x2 * c + x1 * sn;
}

// ---------------------------------------------------------------------------
// Kernel 3: causal GQA flash attention, fp32 WMMA 16x16x4.
// grid = (S/16, 32 heads, B); block = 32 (one wave) per 16-row Q tile.
// K tiles double-buffered via async global->LDS loads; online softmax with
// half-wave shfl_xor row reductions on the WMMA C/D lane layout.
// ---------------------------------------------------------------------------
#define LQS 132     // 128 + 4 pad : conflict-free b64 reads, float4-aligned rows
#define LPS 20      // 16 + 4 pad
__global__ __launch_bounds__(32) void attn_kernel(
    const float* __restrict__ qkv, float* __restrict__ out) {
  __shared__ float lds_q[16 * LQS];
  __shared__ float lds_k[2][16 * LQS];
  __shared__ float lds_p[16 * LPS];

  const int qt   = blockIdx.x;          // q tile (16 rows)
  const int h    = blockIdx.y;
  const int b    = blockIdx.z;
  const int lane = threadIdx.x;
  const int kvh  = h >> 4;              // rep = 32/2 = 16

  const float scale = 0.08838834764831845f;  // 1/sqrt(128)
  const size_t rs = NQKV;

  const float* qbase = qkv + (size_t)(b * SS + qt * 16) * rs + h * HD;
  const float* kbase = qkv + (size_t)(b * SS) * rs + NUM_HEADS * HD + kvh * HD;
  const float* vbase = kbase + NUM_KV * HD;  // V block sits 256 floats after K

  const unsigned int kb0 = (unsigned int)(uintptr_t)&lds_k[0][0];
  const unsigned int kb1 = (unsigned int)(uintptr_t)&lds_k[1][0];

  // async stage of one K tile (16 x 128): 16 b128 issues per lane-slice
  auto issue_k = [&](int kt, unsigned int lbase) {
    const float* kp = kbase + (size_t)(kt * 16) * rs;
#pragma unroll
    for (int i = 0; i < 16; ++i) {
      int f  = lane + i * 32;   // float4 index, 512 total
      int r  = f >> 5;          // 32 float4 per row
      int c4 = f & 31;
      async_ld_b128(lbase + (unsigned int)(r * LQS + c4 * 4) * 4u,
                    &kp[(size_t)r * rs + c4 * 4]);
    }
  };

  issue_k(0, kb0);

  // stage Q tile (16 x 128) coalesced (one-time, synchronous)
#pragma unroll
  for (int i = 0; i < 16; ++i) {
    int f  = lane + i * 32;
    int r  = f >> 5;
    int c4 = f & 31;
    *reinterpret_cast<float4*>(&lds_q[r * LQS + c4 * 4]) =
        *reinterpret_cast<const float4*>(&qbase[(size_t)r * rs + c4 * 4]);
  }
  __syncthreads();

  const int col     = lane & 15;        // A-row / B-col / D-col index
  const int koff    = (lane >> 4) << 1; // K offset 0 or 2
  const int row_off = (lane >> 4) * 8;  // D rows 0-7 vs 8-15

  float m_i[8], l_i[8];
  v8f acc[8] = {};
#pragma unroll
  for (int j = 0; j < 8; ++j) { m_i[j] = -1e30f; l_i[j] = 0.0f; }

  for (int kt = 0; kt <= qt; ++kt) {
    const int cur = kt & 1;
    if (kt < qt) {
      issue_k(kt + 1, cur ? kb0 : kb1);
      asm volatile("s_wait_asynccnt 16" ::: "memory");  // tile kt landed
    } else {
      asm volatile("s_wait_asynccnt 0" ::: "memory");
    }

    // S = Q @ K^T  (K-dim 128 => 32 wmma steps)
    v8f s = {};
#pragma unroll
    for (int kk = 0; kk < 32; ++kk) {
      const int kc = 4 * kk + koff;
      v2f a  = *reinterpret_cast<const v2f*>(&lds_q[col * LQS + kc]);
      v2f bb = *reinterpret_cast<const v2f*>(&lds_k[cur][col * LQS + kc]);
      s = __builtin_amdgcn_wmma_f32_16x16x4_f32(
          false, a, false, bb, (short)0, s, false, false);
    }

    // scale + causal mask + online softmax (per-row stats replicated
    // across the owning half-wave's 16 lanes)
    const int kv_col = kt * 16 + col;
    float p[8];
#pragma unroll
    for (int j = 0; j < 8; ++j) {
      const int qrow = qt * 16 + row_off + j;
      float sv = s[j] * scale;
      if (kv_col > qrow) sv = -1e30f;
      float mx = sv;
      mx = fmaxf(mx, __shfl_xor(mx, 1, 32));
      mx = fmaxf(mx, __shfl_xor(mx, 2, 32));
      mx = fmaxf(mx, __shfl_xor(mx, 4, 32));
      mx = fmaxf(mx, __shfl_xor(mx, 8, 32));
      const float mnew = fmaxf(m_i[j], mx);
      const float pv = __expf(sv - mnew);
      float rsum = pv;
      rsum += __shfl_xor(rsum, 1, 32);
      rsum += __shfl_xor(rsum, 2, 32);
      rsum += __shfl_xor(rsum, 4, 32);
      rsum += __shfl_xor(rsum, 8, 32);
      const float alpha = __expf(m_i[j] - mnew);
      l_i[j] = alpha * l_i[j] + rsum;
      m_i[j] = mnew;
      p[j] = pv;
#pragma unroll
      for (int nt = 0; nt < 8; ++nt) acc[nt][j] *= alpha;
    }

    // transpose P (D layout -> A layout) through LDS
    __syncthreads();
#pragma unroll
    for (int j = 0; j < 8; ++j) lds_p[(row_off + j) * LPS + col] = p[j];
    __syncthreads();

    // O += P @ V   (8 col-tiles x 4 wmma steps); V read directly (64B/half-wave)
    const float* vp = vbase + (size_t)(kt * 16) * rs;
#pragma unroll
    for (int nt = 0; nt < 8; ++nt) {
#pragma unroll
      for (int kk = 0; kk < 4; ++kk) {
        const int kr = 4 * kk + koff;
        v2f a = *reinterpret_cast<const v2f*>(&lds_p[col * LPS + kr]);
        v2f bb;
        bb.x = vp[(size_t)(kr    ) * rs + nt * 16 + col];
        bb.y = vp[(size_t)(kr + 1) * rs + nt * 16 + col];
        acc[nt] = __builtin_amdgcn_wmma_f32_16x16x4_f32(
            false, a, false, bb, (short)0, acc[nt], false, false);
      }
    }
    __syncthreads();
  }

  // epilogue: out[b][qrow][h*128 + c] = acc / l
  float* ob = out + (size_t)(b * SS + qt * 16) * (NUM_HEADS * HD) + h * HD;
#pragma unroll
  for (int j = 0; j < 8; ++j) {
    const float inv_l = 1.0f / l_i[j];
#pragma unroll
    for (int nt = 0; nt < 8; ++nt)
      ob[(size_t)(row_off + j) * (NUM_HEADS * HD) + nt * 16 + col] =
          acc[nt][j] * inv_l;
  }
}

// ---------------------------------------------------------------------------
extern "C" void kernel_launch(void* const* d_in, const int* in_sizes, int n_in,
                              void* d_out, int out_size, void* d_ws, size_t ws_size,
                              hipStream_t stream) {
  const int*   positions = (const int*)d_in[0];
  const float* hidden    = (const float*)d_in[1];
  const float* w_qkv     = (const float*)d_in[2];
  float*       out       = (float*)d_out;
  float*       qkv       = (float*)d_ws;   // B*S*4608 fp32 = ~72 MB scratch

  dim3 g1(HIDDEN / 64, NQKV / 64);         // M=B*S=4096 rows, N=4608 cols
  qkv_gemm_kernel<<<g1, dim3(128), 0, stream>>>(hidden, w_qkv, qkv);

  dim3 g2(BB * SS, NUM_HEADS + 2 * NUM_KV - 2);  // 34 roped heads (32 Q + 2 K)
  rope_kernel<<<g2, dim3(64), 0, stream>>>(positions, qkv);

  dim3 g3(SS / 16, NUM_HEADS, BB);
  attn_kernel<<<g3, dim3(32), 0, stream>>>(qkv, out);
}